// MaskRCNN_37709812859499
// MI455X (gfx1250) — compile-verified
//
#include <hip/hip_runtime.h>
#include <hip/hip_bf16.h>
#include <math.h>

typedef __attribute__((ext_vector_type(2))) float v2f;
typedef __attribute__((ext_vector_type(8))) float v8f;

#define K_PRE   2000
#define K_POST  512
#define NMS_W   63              // ceil(2000/32)
#define IMG_MAX 800.0f
#define CLIP_V  4.135166556742356f
#define LIST_CAP 4096

// ---------- helpers ----------
__device__ inline unsigned score_key(float f) {
    unsigned u = __float_as_uint(f);
    u = (u & 0x80000000u) ? ~u : (u | 0x80000000u);  // monotone ascending
    return ~u;                                       // smaller key = larger score
}

// ---------- 1. init scratch ----------
__global__ void init_kernel(unsigned* __restrict__ hist, unsigned* __restrict__ sel) {
    int i = blockIdx.x * blockDim.x + threadIdx.x;
    if (i < 4096) hist[i] = 0u;
    if (i == 0) { sel[0] = 0u; sel[1] = 0u; }
}

// ---------- 2. decode boxes ----------
__global__ void decode_kernel(const float* __restrict__ anchors,
                              const float* __restrict__ deltas,
                              float* __restrict__ boxes, int N) {
    int i = blockIdx.x * blockDim.x + threadIdx.x;
    if (i >= N) return;
    float a0 = anchors[i*4+0], a1 = anchors[i*4+1], a2 = anchors[i*4+2], a3 = anchors[i*4+3];
    float w = a2 - a0, h = a3 - a1;
    float cx = a0 + 0.5f*w, cy = a1 + 0.5f*h;
    float dx = deltas[i*4+0], dy = deltas[i*4+1];
    float dw = fminf(deltas[i*4+2], CLIP_V);
    float dh = fminf(deltas[i*4+3], CLIP_V);
    float pcx = dx*w + cx, pcy = dy*h + cy;
    float pw = expf(dw)*w, ph = expf(dh)*h;
    float b0 = pcx - 0.5f*pw, b1 = pcy - 0.5f*ph, b2 = pcx + 0.5f*pw, b3 = pcy + 0.5f*ph;
    boxes[i*4+0] = fminf(fmaxf(b0, 0.0f), IMG_MAX);
    boxes[i*4+1] = fminf(fmaxf(b1, 0.0f), IMG_MAX);
    boxes[i*4+2] = fminf(fmaxf(b2, 0.0f), IMG_MAX);
    boxes[i*4+3] = fminf(fmaxf(b3, 0.0f), IMG_MAX);
}

// ---------- 3. score histogram (12-bit, descending buckets) ----------
__global__ void hist_kernel(const float* __restrict__ obj, unsigned* __restrict__ hist, int N) {
    int i = blockIdx.x * blockDim.x + threadIdx.x;
    if (i >= N) return;
    atomicAdd(&hist[score_key(obj[i]) >> 20], 1u);
}

// ---------- 4. find threshold bucket ----------
__global__ void thresh_kernel(const unsigned* __restrict__ hist, unsigned* __restrict__ sel) {
    if (threadIdx.x != 0 || blockIdx.x != 0) return;
    unsigned cum = 0; int b = 0;
    for (; b < 4096; ++b) { cum += hist[b]; if (cum >= K_PRE) break; }
    sel[0] = (unsigned)b;
}

// ---------- 5. compact candidates ----------
__global__ void compact_kernel(const float* __restrict__ obj, unsigned* __restrict__ sel,
                               unsigned long long* __restrict__ list, int N) {
    int i = blockIdx.x * blockDim.x + threadIdx.x;
    if (i >= N) return;
    unsigned key = score_key(obj[i]);
    if ((key >> 20) <= sel[0]) {
        unsigned pos = atomicAdd(&sel[1], 1u);
        if (pos < LIST_CAP) list[pos] = (((unsigned long long)key) << 32) | (unsigned)i;
    }
}

// ---------- 6. bitonic sort 4096 keys, emit top-2000 indices ----------
__global__ __launch_bounds__(1024) void sort_kernel(const unsigned long long* __restrict__ list,
                                                    const unsigned* __restrict__ sel,
                                                    unsigned* __restrict__ sidx) {
    __shared__ unsigned long long sk[LIST_CAP];
    int tid = threadIdx.x;
    unsigned cnt = sel[1]; if (cnt > LIST_CAP) cnt = LIST_CAP;
    for (int i = tid; i < LIST_CAP; i += 1024)
        sk[i] = (i < (int)cnt) ? list[i] : 0xFFFFFFFFFFFFFFFFull;
    __syncthreads();
    for (int k = 2; k <= LIST_CAP; k <<= 1) {
        for (int j = k >> 1; j > 0; j >>= 1) {
            for (int i = tid; i < LIST_CAP; i += 1024) {
                int l = i ^ j;
                if (l > i) {
                    unsigned long long a = sk[i], b = sk[l];
                    bool up = ((i & k) == 0);
                    if ((a > b) == up) { sk[i] = b; sk[l] = a; }
                }
            }
            __syncthreads();
        }
    }
    for (int i = tid; i < K_PRE; i += 1024) sidx[i] = (unsigned)(sk[i] & 0xFFFFFFFFull);
}

// ---------- 7. gather candidate boxes + areas ----------
__global__ void gather_kernel(const float* __restrict__ boxes, const unsigned* __restrict__ sidx,
                              float* __restrict__ cand, float* __restrict__ area) {
    int j = blockIdx.x * blockDim.x + threadIdx.x;
    if (j >= K_PRE) return;
    unsigned gi = sidx[j];
    float b0 = boxes[gi*4+0], b1 = boxes[gi*4+1], b2 = boxes[gi*4+2], b3 = boxes[gi*4+3];
    cand[j*4+0] = b0; cand[j*4+1] = b1; cand[j*4+2] = b2; cand[j*4+3] = b3;
    area[j] = (b2 - b0) * (b3 - b1);
}

// ---------- 8. NMS suppression bitmask ----------
__global__ void mask_kernel(const float* __restrict__ cand, const float* __restrict__ area,
                            unsigned* __restrict__ masks) {
    int t = blockIdx.x * blockDim.x + threadIdx.x;
    if (t >= K_PRE * NMS_W) return;
    int i = t / NMS_W, w = t - i * NMS_W;
    float ax1 = cand[i*4+0], ay1 = cand[i*4+1], ax2 = cand[i*4+2], ay2 = cand[i*4+3];
    float ar = area[i];
    unsigned m = 0;
    int jbase = w * 32;
    #pragma unroll 4
    for (int b = 0; b < 32; ++b) {
        int j = jbase + b;
        if (j < K_PRE && j > i) {
            float bx1 = cand[j*4+0], by1 = cand[j*4+1], bx2 = cand[j*4+2], by2 = cand[j*4+3];
            float ltx = fmaxf(ax1, bx1), lty = fmaxf(ay1, by1);
            float rbx = fminf(ax2, bx2), rby = fminf(ay2, by2);
            float iw = fmaxf(rbx - ltx, 0.0f), ih = fmaxf(rby - lty, 0.0f);
            float inter = iw * ih;
            float iou = inter / (ar + area[j] - inter + 1e-6f);
            if (iou > 0.7f) m |= (1u << b);
        }
    }
    masks[i * NMS_W + w] = m;
}

// ---------- 9. sequential resolve ----------
__global__ __launch_bounds__(64) void resolve_kernel(const unsigned* __restrict__ masks,
                                                     unsigned* __restrict__ keep) {
    __shared__ unsigned supp[NMS_W];
    int tid = threadIdx.x;
    if (tid < NMS_W) supp[tid] = 0u;
    __syncthreads();
    for (int i = 0; i < K_PRE; ++i) {
        unsigned word = supp[i >> 5];
        bool suppressed = (word >> (i & 31)) & 1u;
        if (!suppressed && tid < NMS_W) atomicOr(&supp[tid], masks[i * NMS_W + tid]);
        __syncthreads();
    }
    for (int j = tid; j < K_PRE; j += 64)
        keep[j] = 1u - ((supp[j >> 5] >> (j & 31)) & 1u);
}

// ---------- 10. final selection (order-preserving compaction) ----------
__global__ void select_kernel(const float* __restrict__ cand, const unsigned* __restrict__ keep,
                              float* __restrict__ fb, unsigned* __restrict__ valid) {
    if (threadIdx.x != 0 || blockIdx.x != 0) return;
    int cnt = 0;
    for (int i = 0; i < K_PRE && cnt < K_POST; ++i) {
        if (keep[i]) {
            fb[cnt*4+0] = cand[i*4+0]; fb[cnt*4+1] = cand[i*4+1];
            fb[cnt*4+2] = cand[i*4+2]; fb[cnt*4+3] = cand[i*4+3];
            valid[cnt] = 1u; ++cnt;
        }
    }
    for (; cnt < K_POST; ++cnt) valid[cnt] = 0u;
}

// ---------- 11. zero output ----------
__global__ void zero_kernel(float* __restrict__ out, int total) {
    int i = blockIdx.x * blockDim.x + threadIdx.x;
    if (i < total) out[i] = 0.0f;
}

// ---------- 12. ROI-align + WMMA avg-pool ----------
// One workgroup per box (256 threads = 8 waves). LDS sample layout s' = p*4 + d,
// so each K=4 block of V_WMMA_F32_16X16X4_F32 is exactly one pool cell's 4 samples.
// Wave-uniform loop bounds are pinned to SGPRs via readfirstlane so the backend
// emits scalar loop control and EXEC stays all-ones across the WMMA clause
// (ISA 7.12 requirement).
__global__ __launch_bounds__(256) void roi_kernel(const float* __restrict__ feat,
                                                  const float* __restrict__ fb,
                                                  const unsigned* __restrict__ valid,
                                                  float* __restrict__ out) {
    int box = blockIdx.x;
    if (!valid[box]) return;
    const int tid = threadIdx.x;
    __shared__ int   so[4][196];
    __shared__ float swt[4][196];
    __shared__ float val[32 * 200];

    float x1 = fb[box*4+0] * 0.25f, y1 = fb[box*4+1] * 0.25f;
    float x2 = fb[box*4+2] * 0.25f, y2 = fb[box*4+3] * 0.25f;
    float rw = fmaxf(x2 - x1, 1.0f), rh = fmaxf(y2 - y1, 1.0f);

    if (tid < 196) {
        int p = tid >> 2, d = tid & 3;
        int py = p / 7, px = p % 7;
        int iy = 2*py + (d >> 1), ix = 2*px + (d & 1);
        float y = y1 + ((float)iy + 0.5f) * (rh * (1.0f/14.0f));
        float x = x1 + ((float)ix + 0.5f) * (rw * (1.0f/14.0f));
        y = fminf(fmaxf(y, 0.0f), 199.0f);
        x = fminf(fmaxf(x, 0.0f), 199.0f);
        int y0 = (int)floorf(y), x0 = (int)floorf(x);
        int y1i = min(y0 + 1, 199), x1i = min(x0 + 1, 199);
        float ly = y - (float)y0, lx = x - (float)x0;
        so[0][tid] = y0 *200 + x0;  swt[0][tid] = (1.0f-ly)*(1.0f-lx);
        so[1][tid] = y0 *200 + x1i; swt[1][tid] = (1.0f-ly)*lx;
        so[2][tid] = y1i*200 + x0;  swt[2][tid] = ly*(1.0f-lx);
        so[3][tid] = y1i*200 + x1i; swt[3][tid] = ly*lx;
    }
    __syncthreads();

    const int wv = tid >> 5, lane = tid & 31;
    // wave-uniform role constants, pinned to SGPRs
    const int chalf = __builtin_amdgcn_readfirstlane(wv >> 2);
    const int n0    = __builtin_amdgcn_readfirstlane((wv & 3) * 16);
    const int kend  = __builtin_amdgcn_readfirstlane((n0 + 16 < 49) ? n0 + 16 : 49);
    const int crow  = chalf * 16 + (lane & 15);
    const int koff  = (lane < 16) ? 0 : 2;

    for (int h = 0; h < 8; ++h) {           // 8 x 32 channels
        for (int idx = tid; idx < 32 * 196; idx += 256) {
            int cl = idx / 196, s = idx - cl * 196;
            const float* fp = feat + (size_t)(h*32 + cl) * 40000;
            float v = swt[0][s]*fp[so[0][s]] + swt[1][s]*fp[so[1][s]]
                    + swt[2][s]*fp[so[2][s]] + swt[3][s]*fp[so[3][s]];
            val[cl * 200 + s] = v;
        }
        __syncthreads();

        v8f acc = {0.f,0.f,0.f,0.f,0.f,0.f,0.f,0.f};
        for (int k = n0; k < kend; ++k) {
            v2f a; a.x = val[crow*200 + 4*k + koff]; a.y = val[crow*200 + 4*k + koff + 1];
            float bw = ((lane & 15) == (k - n0)) ? 0.25f : 0.0f;
            v2f b; b.x = bw; b.y = bw;
            acc = __builtin_amdgcn_wmma_f32_16x16x4_f32(
                false, a, false, b, (short)0, acc, false, false);
        }

        int n = lane & 15;
        int p = n0 + n;
        size_t obase = ((size_t)box * 256 + (size_t)(h*32 + chalf*16)) * 49;
        if (p < 49) {
            #pragma unroll
            for (int g = 0; g < 8; ++g) {
                int M = (lane < 16) ? g : (g + 8);
                out[obase + (size_t)M * 49 + p] = acc[g];
            }
        }
        __syncthreads();
    }
}

// ---------- launch ----------
extern "C" void kernel_launch(void* const* d_in, const int* in_sizes, int n_in,
                              void* d_out, int out_size, void* d_ws, size_t ws_size,
                              hipStream_t stream) {
    const float* feature    = (const float*)d_in[0];
    const float* objectness = (const float*)d_in[1];
    const float* deltas     = (const float*)d_in[2];
    const float* anchors    = (const float*)d_in[3];
    const int N = in_sizes[1];

    char* ws = (char*)d_ws;
    size_t off = 0;
    auto take = [&](size_t bytes) { size_t cur = off; off = (off + bytes + 255) & ~(size_t)255; return cur; };
    float*              boxes = (float*)             (ws + take((size_t)N * 4 * sizeof(float)));
    unsigned*           hist  = (unsigned*)          (ws + take(4096 * sizeof(unsigned)));
    unsigned*           sel   = (unsigned*)          (ws + take(4 * sizeof(unsigned)));
    unsigned long long* list  = (unsigned long long*)(ws + take(LIST_CAP * sizeof(unsigned long long)));
    unsigned*           sidx  = (unsigned*)          (ws + take(2048 * sizeof(unsigned)));
    float*              cand  = (float*)             (ws + take(K_PRE * 4 * sizeof(float)));
    float*              area  = (float*)             (ws + take(K_PRE * sizeof(float)));
    unsigned*           keep  = (unsigned*)          (ws + take(K_PRE * sizeof(unsigned)));
    unsigned*           masks = (unsigned*)          (ws + take((size_t)K_PRE * NMS_W * sizeof(unsigned)));
    float*              fb    = (float*)             (ws + take(K_POST * 4 * sizeof(float)));
    unsigned*           valid = (unsigned*)          (ws + take(K_POST * sizeof(unsigned)));
    float* out = (float*)d_out;

    int nb = (N + 255) / 256;
    init_kernel   <<<(4096 + 255) / 256, 256, 0, stream>>>(hist, sel);
    decode_kernel <<<nb, 256, 0, stream>>>(anchors, deltas, boxes, N);
    hist_kernel   <<<nb, 256, 0, stream>>>(objectness, hist, N);
    thresh_kernel <<<1, 1, 0, stream>>>(hist, sel);
    compact_kernel<<<nb, 256, 0, stream>>>(objectness, sel, list, N);
    sort_kernel   <<<1, 1024, 0, stream>>>(list, sel, sidx);
    gather_kernel <<<(K_PRE + 255) / 256, 256, 0, stream>>>(boxes, sidx, cand, area);
    mask_kernel   <<<(K_PRE * NMS_W + 255) / 256, 256, 0, stream>>>(cand, area, masks);
    resolve_kernel<<<1, 64, 0, stream>>>(masks, keep);
    select_kernel <<<1, 1, 0, stream>>>(cand, keep, fb, valid);
    zero_kernel   <<<(out_size + 1023) / 1024, 1024, 0, stream>>>(out, out_size);
    roi_kernel    <<<K_POST, 256, 0, stream>>>(feature, fb, valid, out);
}